// YOLOv1_40553081208874
// MI455X (gfx1250) — compile-verified
//
#include <hip/hip_runtime.h>
#include <hip/hip_bf16.h>
#include <math.h>

typedef float v2f __attribute__((ext_vector_type(2)));
typedef float v8f __attribute__((ext_vector_type(8)));

#define Bsz    32
#define Csz    512
#define Wimg   40
#define Ncell  1600          // 40*40
#define NCLS   20
#define STRIDE_F 32.0f
#define CONF_T 0.3f
#define NMS_T  0.5f
#define NP     2048          // NMS sort padding (pow2 >= 1600)

#ifdef __has_builtin
#  if __has_builtin(__builtin_amdgcn_global_load_async_to_lds_b128)
#    define USE_ASYNC_LDS 1
#  endif
#  if __has_builtin(__builtin_amdgcn_s_wait_asynccnt)
#    define HAVE_WAIT_ASYNC 1
#  endif
#endif

__device__ __forceinline__ float sigmoidf_(float x) { return 1.0f / (1.0f + __expf(-x)); }

// ---------------------------------------------------------------------------
// Kernel 0: pack [25,512] head weights into parity-interleaved Wt:
//   element (channel k, output o) -> wt[(k>>1)*64 + o*2 + (k&1)]
// so the (k, k+1) pair for one output column is an aligned 8-byte unit.
// o: 0=obj, 1..20=cls, 21..24=reg, 25..31=zero
// ---------------------------------------------------------------------------
__global__ __launch_bounds__(256) void pack_weights(
    const float* __restrict__ wobj, const float* __restrict__ wcls,
    const float* __restrict__ wreg, float* __restrict__ wt)
{
    int i = blockIdx.x * blockDim.x + threadIdx.x;   // exactly 512*32 threads
    int k = i >> 5;
    int o = i & 31;
    float v = 0.0f;
    if (o == 0)       v = wobj[k];
    else if (o <= 20) v = wcls[(o - 1) * Csz + k];
    else if (o <= 24) v = wreg[(o - 21) * Csz + k];
    wt[(k >> 1) * 64 + o * 2 + (k & 1)] = v;
}

// ---------------------------------------------------------------------------
// Kernel 1: head GEMM via V_WMMA_F32_16X16X4_F32 + box decode
// One wave per 16-cell tile. 3200 tiles total -> 400 blocks x 8 waves.
// Weights staged once per block into LDS (async-to-LDS when available);
// A-stream software-pipelined; B fragments are single ds_load_b64 each.
// ---------------------------------------------------------------------------
__global__ __launch_bounds__(256) void head_gemm_decode(
    const float* __restrict__ feat, const float* __restrict__ wt,
    const float* __restrict__ b_obj, const float* __restrict__ b_cls,
    const float* __restrict__ b_reg, float* __restrict__ out)
{
    __shared__ float wlds[Csz * 32];                  // 64 KB; reused as hbuf later

    const int lane = threadIdx.x & 31;
    const int wave = threadIdx.x >> 5;
    const int tile = blockIdx.x * 8 + wave;           // 0..3199
    const int b    = tile / 100;                      // 100 tiles per image
    const int n0   = (tile % 100) * 16;
    const int m    = lane & 15;
    const int ksel = (lane >> 4) << 1;                // 0 or 2 (K / K+2 halves)

    // ---- stage weights to LDS --------------------------------------------
#ifdef USE_ASYNC_LDS
    {
        typedef int v4i_ __attribute__((vector_size(16)));
        typedef __attribute__((address_space(1))) v4i_ gv4i;
        typedef __attribute__((address_space(3))) v4i_ lv4i;
        for (int t = threadIdx.x; t < (Csz * 32 / 4); t += 256) {
            __builtin_amdgcn_global_load_async_to_lds_b128(
                (gv4i*)wt + t, (lv4i*)wlds + t, 0, 0);
        }
#  ifdef HAVE_WAIT_ASYNC
        __builtin_amdgcn_s_wait_asynccnt(0);
#  else
        asm volatile("s_wait_asynccnt 0x0" ::: "memory");
#  endif
    }
#else
    for (int t = threadIdx.x; t < (Csz * 32 / 4); t += 256)
        ((float4*)wlds)[t] = ((const float4*)wt)[t];
#endif
    __syncthreads();

    // ---- K loop -----------------------------------------------------------
    const float* fbase = feat + (size_t)b * (Csz * Ncell) + n0 + m;
    const int    col   = lane & 15;
    const float* wb    = wlds + col * 2;              // interleaved layout base

    v8f acc0 = {0.f, 0.f, 0.f, 0.f, 0.f, 0.f, 0.f, 0.f};
    v8f acc1 = {0.f, 0.f, 0.f, 0.f, 0.f, 0.f, 0.f, 0.f};

    v2f a;
    a.x = fbase[(size_t)(ksel)     * Ncell];
    a.y = fbase[(size_t)(ksel + 1) * Ncell];

#pragma unroll 4
    for (int k = 0; k < Csz - 4; k += 4) {
        // software-pipelined A stream: issue next loads before consuming `a`
        v2f a_n;
        a_n.x = fbase[(size_t)(k + 4 + ksel)     * Ncell];
        a_n.y = fbase[(size_t)(k + 4 + ksel + 1) * Ncell];
        __builtin_prefetch(fbase + (size_t)(k + 20 + ksel) * Ncell, 0, 1);

        const int p = (k + ksel) >> 1;                // K-pair index
        v2f bb0 = *(const v2f*)(wb + p * 64);         // cols 0..15
        v2f bb1 = *(const v2f*)(wb + p * 64 + 32);    // cols 16..31
        acc0 = __builtin_amdgcn_wmma_f32_16x16x4_f32(false, a, false, bb0,
                                                     (short)0, acc0, false, false);
        acc1 = __builtin_amdgcn_wmma_f32_16x16x4_f32(false, a, false, bb1,
                                                     (short)0, acc1, false, false);
        a = a_n;
    }
    {   // peeled last K-step (k = Csz-4), consumes the in-flight fragment
        const int p = (Csz - 4 + ksel) >> 1;
        v2f bb0 = *(const v2f*)(wb + p * 64);
        v2f bb1 = *(const v2f*)(wb + p * 64 + 32);
        acc0 = __builtin_amdgcn_wmma_f32_16x16x4_f32(false, a, false, bb0,
                                                     (short)0, acc0, false, false);
        acc1 = __builtin_amdgcn_wmma_f32_16x16x4_f32(false, a, false, bb1,
                                                     (short)0, acc1, false, false);
    }

    // ---- epilogue: regroup per-cell via LDS (reuse weight region) ---------
    __syncthreads();                                  // everyone done reading weights
    float* hb = wlds + wave * (16 * 32);
    const int rbase = (lane >> 4) * 8;
#pragma unroll
    for (int v = 0; v < 8; ++v) {
        hb[(rbase + v) * 32 + col]      = acc0[v];
        hb[(rbase + v) * 32 + col + 16] = acc1[v];
    }
    __syncthreads();

    if (lane < 16) {
        const int n = n0 + lane;                      // cell index within image
        const float* row = hb + lane * 32;

        float sobj = sigmoidf_(row[0] + b_obj[0]);
        float best = -1.0f;
        int bestc = 0;
#pragma unroll
        for (int c = 0; c < NCLS; ++c) {
            float s = sqrtf(sobj * sigmoidf_(row[1 + c] + b_cls[c]));
            if (s > best) { best = s; bestc = c; }    // strict > keeps first max
        }

        float r0 = row[21] + b_reg[0];
        float r1 = row[22] + b_reg[1];
        float r2 = row[23] + b_reg[2];
        float r3 = row[24] + b_reg[3];
        float gx = (float)(n % Wimg);
        float gy = (float)(n / Wimg);
        float cx = (sigmoidf_(r0) + gx) * STRIDE_F;
        float cy = (sigmoidf_(r1) + gy) * STRIDE_F;
        float bw = __expf(r2) * STRIDE_F;
        float bh = __expf(r3) * STRIDE_F;

        size_t cell = (size_t)b * Ncell + n;
        float* o5 = out + cell * 5;
        o5[0] = cx - bw * 0.5f;
        o5[1] = cy - bh * 0.5f;
        o5[2] = cx + bw * 0.5f;
        o5[3] = cy + bh * 0.5f;
        o5[4] = best;
        out[(size_t)Bsz * Ncell * 5 + cell] = (float)bestc;   // labels section
    }
}

// ---------------------------------------------------------------------------
// Kernel 2: per-image greedy NMS. One 256-thread workgroup per image.
// Bitonic sort (score desc, idx tiebreak) in LDS, greedy suppression loop,
// scatter keep back to original order, zero non-kept rows.
// ---------------------------------------------------------------------------
__global__ __launch_bounds__(256) void nms_kernel(float* __restrict__ out)
{
    __shared__ float skey[NP];
    __shared__ int   sidx[NP];
    __shared__ float sx1[NP], sy1[NP], sx2[NP], sy2[NP];
    __shared__ int   slab[NP];
    __shared__ unsigned char skeep[NP];

    const int b   = blockIdx.x;
    const int tid = threadIdx.x;
    const int nt  = blockDim.x;

    float* osec = out;                                // [B,N,5]
    float* lsec = out + (size_t)Bsz * Ncell * 5;      // labels [B,N]
    float* ksec = lsec + (size_t)Bsz * Ncell;         // keep   [B,N]

    for (int i = tid; i < NP; i += nt) {
        if (i < Ncell) {
            skey[i] = osec[((size_t)b * Ncell + i) * 5 + 4];
            sidx[i] = i;
        } else {
            skey[i] = -1e30f;
            sidx[i] = i;                              // >= Ncell marks padding
        }
    }
    __syncthreads();

    // Bitonic sort, descending by key, ascending idx on ties (deterministic).
    for (int k = 2; k <= NP; k <<= 1) {
        for (int j = k >> 1; j > 0; j >>= 1) {
            for (int i = tid; i < NP; i += nt) {
                int ixj = i ^ j;
                if (ixj > i) {
                    float ka = skey[i], kb = skey[ixj];
                    int   ia = sidx[i], ib = sidx[ixj];
                    bool before = (ka > kb) || (ka == kb && ia < ib);
                    bool desc   = ((i & k) == 0);
                    if (desc ? !before : before) {
                        skey[i] = kb; skey[ixj] = ka;
                        sidx[i] = ib; sidx[ixj] = ia;
                    }
                }
            }
            __syncthreads();
        }
    }

    // Gather boxes/labels into sorted order; init keep = conf mask.
    for (int i = tid; i < NP; i += nt) {
        int orig = sidx[i];
        if (orig < Ncell) {
            const float* o5 = osec + ((size_t)b * Ncell + orig) * 5;
            sx1[i] = o5[0]; sy1[i] = o5[1]; sx2[i] = o5[2]; sy2[i] = o5[3];
            slab[i] = (int)lsec[(size_t)b * Ncell + orig];
            skeep[i] = (skey[i] > CONF_T) ? 1 : 0;
        } else {
            sx1[i] = 0.f; sy1[i] = 0.f; sx2[i] = 0.f; sy2[i] = 0.f;
            slab[i] = -1;
            skeep[i] = 0;
        }
    }
    __syncthreads();

    // Greedy: if box i (sorted) is kept, suppress later same-class overlaps.
    for (int i = 0; i < Ncell; ++i) {
        if (skeep[i]) {
            float ax1 = sx1[i], ay1 = sy1[i], ax2 = sx2[i], ay2 = sy2[i];
            float aarea = (ax2 - ax1) * (ay2 - ay1);
            int   al = slab[i];
            for (int j = i + 1 + tid; j < Ncell; j += nt) {
                if (skeep[j] && slab[j] == al) {
                    float x1 = fmaxf(ax1, sx1[j]);
                    float y1 = fmaxf(ay1, sy1[j]);
                    float x2 = fminf(ax2, sx2[j]);
                    float y2 = fminf(ay2, sy2[j]);
                    float inter = fmaxf(x2 - x1, 0.f) * fmaxf(y2 - y1, 0.f);
                    float barea = (sx2[j] - sx1[j]) * (sy2[j] - sy1[j]);
                    float iou = inter / (aarea + barea - inter);
                    if (iou > NMS_T) skeep[j] = 0;
                }
            }
        }
        __syncthreads();
    }

    // Scatter keep to original order; zero non-kept rows of the output.
    for (int i = tid; i < NP; i += nt) {
        int orig = sidx[i];
        if (orig < Ncell) {
            size_t cell = (size_t)b * Ncell + orig;
            ksec[cell] = skeep[i] ? 1.0f : 0.0f;
            if (!skeep[i]) {
                float* o5 = osec + cell * 5;
                o5[0] = 0.f; o5[1] = 0.f; o5[2] = 0.f; o5[3] = 0.f; o5[4] = 0.f;
            }
        }
    }
}

// ---------------------------------------------------------------------------
extern "C" void kernel_launch(void* const* d_in, const int* in_sizes, int n_in,
                              void* d_out, int out_size, void* d_ws, size_t ws_size,
                              hipStream_t stream)
{
    const float* feat = (const float*)d_in[0];
    const float* wobj = (const float*)d_in[1];
    const float* bobj = (const float*)d_in[2];
    const float* wcls = (const float*)d_in[3];
    const float* bcls = (const float*)d_in[4];
    const float* wreg = (const float*)d_in[5];
    const float* breg = (const float*)d_in[6];
    float* out = (float*)d_out;
    float* wt  = (float*)d_ws;                        // interleaved Wt = 64 KB

    pack_weights<<<64, 256, 0, stream>>>(wobj, wcls, wreg, wt);
    head_gemm_decode<<<400, 256, 0, stream>>>(feat, wt, bobj, bcls, breg, out);
    nms_kernel<<<Bsz, 256, 0, stream>>>(out);
}